// TC_81501299409222
// MI455X (gfx1250) — compile-verified
//
#include <hip/hip_runtime.h>
#include <hip/hip_bf16.h>
#include <math.h>

// ---------------------------------------------------------------------------
// Problem constants (match reference)
// ---------------------------------------------------------------------------
#define Bdim 512
#define Cdim 256
#define Ldim 256
#define Hdim 512
#define Tdim 16
#define Wdim 8
#define BW   (Bdim * Wdim)          // 4096
#define C2   (Cdim / 2)             // 128
#define C4   (Cdim / 4)             // 64

typedef float v2f __attribute__((ext_vector_type(2)));
typedef float v8f __attribute__((ext_vector_type(8)));

// ---------------------------------------------------------------------------
// WMMA f32 16x16x4 fragment helpers (wave32).
// A (16x4) / B-as-rows (16xK, N in lanes) fragment: lane l<16 holds row l,
// K = {k, k+1}; lane l>=16 holds row l-16, K = {k+2, k+3}.
// C/D: VGPR r, lane l -> row r + 8*(l>>4), col l&15.
// ---------------------------------------------------------------------------
__device__ __forceinline__ v2f frag_rowmajor(const float* __restrict__ base,
                                             int row0, int ld, int k, int lane) {
    const int r  = lane & 15;
    const int kk = k + ((lane >> 4) << 1);
    const float* p = base + (size_t)(row0 + r) * ld + kk;
    v2f v; v.x = p[0]; v.y = p[1];
    return v;
}

// ---------------------------------------------------------------------------
// Kernel 1: pred[t, bw, c] = c_t[bw,:] . Wk_w[t,c,:] + Wk_b[t,c]
// c_t viewed as [4096 x 512] (ld 512), Wk_w[t] as [256 x 512] (ld 512).
// Block: 256 thr = 8 waves as 4x2 -> 64x32 tile. Grid: (64, 8, T).
// ---------------------------------------------------------------------------
__global__ void k1_pred_gemm(const float* __restrict__ c_t,
                             const float* __restrict__ Wk_w,
                             const float* __restrict__ Wk_b,
                             float* __restrict__ pred) {
    const int t    = blockIdx.z;
    const int lane = threadIdx.x & 31;
    const int wid  = threadIdx.x >> 5;
    const int m0   = blockIdx.x * 64 + (wid >> 1) * 16;   // row in [0,4096)
    const int n0   = blockIdx.y * 32 + (wid & 1) * 16;    // col in [0,256)

    const float* A  = c_t;                                // [4096 x 512]
    const float* Bm = Wk_w + (size_t)t * Cdim * Hdim;     // [256 x 512]

    v8f acc = {};
    for (int k = 0; k < Hdim; k += 4) {
        v2f a = frag_rowmajor(A,  m0, Hdim, k, lane);
        v2f b = frag_rowmajor(Bm, n0, Hdim, k, lane);
        acc = __builtin_amdgcn_wmma_f32_16x16x4_f32(false, a, false, b,
                                                    (short)0, acc, false, false);
    }

    const int   col   = n0 + (lane & 15);
    const float bias  = Wk_b[t * Cdim + col];
    const int   rbase = m0 + ((lane >> 4) << 3);
    float* D = pred + (size_t)t * BW * Cdim;              // [4096 x 256]
#pragma unroll
    for (int r = 0; r < 8; ++r)
        D[(size_t)(rbase + r) * Cdim + col] = acc[r] + bias;
}

// ---------------------------------------------------------------------------
// Kernel 2: fused scores + log-softmax diagonal partial sums.
// Block handles (t, w, 16-row block b0..b0+15) and ALL 512 columns.
// encode tile (strided gather from features_aug2) staged in LDS; each of the
// 8 waves computes 4 16x16 tiles (64 columns); full 16x512 slab reduced in
// LDS deterministically. Grid: (32, 8, 16), 256 threads.
// ---------------------------------------------------------------------------
__global__ void k2_scores_nce(const float* __restrict__ feats,
                              const float* __restrict__ pred,
                              const int*   __restrict__ t_sample,
                              float*       __restrict__ partials) {
    __shared__ float Es[16][Cdim];        // 16 KB : encode tile, K contiguous
    __shared__ float Ss[16][Bdim];        // 32 KB : score slab
    __shared__ float red[16][16];
    __shared__ float rowM[16];
    __shared__ float rowS[16];

    const int rb = blockIdx.x, w = blockIdx.y, t = blockIdx.z;
    const int b0 = rb * 16;
    const int tid = threadIdx.x;

    int start = t_sample[0] + 1 + t;
    if (start > Ldim - Wdim) start = Ldim - Wdim;
    const int pos = start + w;

    // gather encode[b0..b0+15, 0..255] : feats[b, c, pos], stride L over c
    for (int i = tid; i < 16 * Cdim; i += 256) {
        const int r = i >> 8, c = i & (Cdim - 1);
        Es[r][c] = feats[(size_t)(b0 + r) * (Cdim * Ldim) + (size_t)c * Ldim + pos];
    }
    __syncthreads();

    const int lane = tid & 31, wid = tid >> 5;
    const float* P = pred + (size_t)t * BW * Cdim + w * Cdim;  // row a: +a*2048

    v8f acc[4] = {};
    const int rA = lane & 15;
    for (int k = 0; k < Cdim; k += 4) {
        const int kA = k + ((lane >> 4) << 1);
        v2f a; a.x = Es[rA][kA]; a.y = Es[rA][kA + 1];
#pragma unroll
        for (int ti = 0; ti < 4; ++ti) {
            const int a0 = wid * 64 + ti * 16;
            const float* p = P + (size_t)(a0 + rA) * (Wdim * Cdim) + kA;
            v2f b; b.x = p[0]; b.y = p[1];
            acc[ti] = __builtin_amdgcn_wmma_f32_16x16x4_f32(false, a, false, b,
                                                            (short)0, acc[ti],
                                                            false, false);
        }
    }

    // spill score slab to LDS
#pragma unroll
    for (int ti = 0; ti < 4; ++ti) {
        const int col   = wid * 64 + ti * 16 + (lane & 15);
        const int rbase = (lane >> 4) << 3;
#pragma unroll
        for (int r = 0; r < 8; ++r) Ss[rbase + r][col] = acc[ti][r];
    }
    __syncthreads();

    // deterministic per-row max / sum-exp: 16 threads per row, 32 cols each
    const int row = tid >> 4, j = tid & 15;
    float m = -3.402823466e38f;
    for (int c = j * 32; c < j * 32 + 32; ++c) m = fmaxf(m, Ss[row][c]);
    red[row][j] = m;
    __syncthreads();
    if (j == 0) {
        float mm = red[row][0];
        for (int q = 1; q < 16; ++q) mm = fmaxf(mm, red[row][q]);
        rowM[row] = mm;
    }
    __syncthreads();
    {
        const float mm = rowM[row];
        float s = 0.f;
        for (int c = j * 32; c < j * 32 + 32; ++c) s += __expf(Ss[row][c] - mm);
        red[row][j] = s;
    }
    __syncthreads();
    if (j == 0) {
        float ss = 0.f;
        for (int q = 0; q < 16; ++q) ss += red[row][q];
        rowS[row] = ss;
    }
    __syncthreads();
    if (tid < 16) {
        const float diag = Ss[tid][b0 + tid];               // logits[b, a=b]
        red[0][tid] = diag - rowM[tid] - __logf(rowS[tid]); // log-softmax diag
    }
    __syncthreads();
    if (tid == 0) {
        float tot = 0.f;
        for (int q = 0; q < 16; ++q) tot += red[0][q];
        partials[(t * Wdim + w) * 32 + rb] = tot;
    }
}

// ---------------------------------------------------------------------------
// Kernel 3: h1[4096 x 128] = c_t2d[4096 x 512] @ pw1^T[512 x 128] + pb1
// Grid: (64, 4). Same 64x32-per-block WMMA scheme.
// ---------------------------------------------------------------------------
__global__ void k3_h1_gemm(const float* __restrict__ c_t,
                           const float* __restrict__ pw1,
                           const float* __restrict__ pb1,
                           float* __restrict__ h1) {
    const int lane = threadIdx.x & 31;
    const int wid  = threadIdx.x >> 5;
    const int m0   = blockIdx.x * 64 + (wid >> 1) * 16;
    const int n0   = blockIdx.y * 32 + (wid & 1) * 16;

    v8f acc = {};
    for (int k = 0; k < Hdim; k += 4) {
        v2f a = frag_rowmajor(c_t, m0, Hdim, k, lane);
        v2f b = frag_rowmajor(pw1, n0, Hdim, k, lane);
        acc = __builtin_amdgcn_wmma_f32_16x16x4_f32(false, a, false, b,
                                                    (short)0, acc, false, false);
    }
    const int   col   = n0 + (lane & 15);
    const float bias  = pb1[col];
    const int   rbase = m0 + ((lane >> 4) << 3);
#pragma unroll
    for (int r = 0; r < 8; ++r)
        h1[(size_t)(rbase + r) * C2 + col] = acc[r] + bias;
}

// ---------------------------------------------------------------------------
// Kernel 4: BatchNorm train stats per column of h1 (4096 rows).
// Produces fused affine: scale = gamma*rsqrt(var+eps), shift = beta - mean*scale
// Grid: (128). 256 threads, deterministic tree reduction.
// ---------------------------------------------------------------------------
__global__ void k4_bn_stats(const float* __restrict__ h1,
                            const float* __restrict__ gamma,
                            const float* __restrict__ beta,
                            float* __restrict__ scaleArr,
                            float* __restrict__ shiftArr) {
    __shared__ float s1[256], s2[256];
    const int c = blockIdx.x, tid = threadIdx.x;
    float a = 0.f, b = 0.f;
    for (int r = tid; r < BW; r += 256) {
        const float v = h1[(size_t)r * C2 + c];
        a += v; b += v * v;
    }
    s1[tid] = a; s2[tid] = b;
    __syncthreads();
    for (int off = 128; off > 0; off >>= 1) {
        if (tid < off) { s1[tid] += s1[tid + off]; s2[tid] += s2[tid + off]; }
        __syncthreads();
    }
    if (tid == 0) {
        const float mean = s1[0] * (1.f / (float)BW);
        const float var  = s2[0] * (1.f / (float)BW) - mean * mean;
        const float sc   = rsqrtf(var + 1e-5f) * gamma[c];
        scaleArr[c] = sc;
        shiftArr[c] = beta[c] - mean * sc;
    }
}

// ---------------------------------------------------------------------------
// Kernel 5: proj[4096 x 64] = relu(bn(h1)) @ pw2^T[128 x 64] + pb2 -> d_out+1
// BN + ReLU applied on-the-fly while building A fragments. Grid: (64, 2).
// ---------------------------------------------------------------------------
__global__ void k5_proj_gemm(const float* __restrict__ h1,
                             const float* __restrict__ scaleArr,
                             const float* __restrict__ shiftArr,
                             const float* __restrict__ pw2,
                             const float* __restrict__ pb2,
                             float* __restrict__ proj) {
    const int lane = threadIdx.x & 31;
    const int wid  = threadIdx.x >> 5;
    const int m0   = blockIdx.x * 64 + (wid >> 1) * 16;
    const int n0   = blockIdx.y * 32 + (wid & 1) * 16;

    const int rA = lane & 15;
    v8f acc = {};
    for (int k = 0; k < C2; k += 4) {
        const int kA = k + ((lane >> 4) << 1);
        const float* hp = h1 + (size_t)(m0 + rA) * C2 + kA;
        v2f a;
        a.x = fmaxf(hp[0] * scaleArr[kA]     + shiftArr[kA],     0.f);
        a.y = fmaxf(hp[1] * scaleArr[kA + 1] + shiftArr[kA + 1], 0.f);
        v2f b = frag_rowmajor(pw2, n0, C2, k, lane);
        acc = __builtin_amdgcn_wmma_f32_16x16x4_f32(false, a, false, b,
                                                    (short)0, acc, false, false);
    }
    const int   col   = n0 + (lane & 15);
    const float bias  = pb2[col];
    const int   rbase = m0 + ((lane >> 4) << 3);
#pragma unroll
    for (int r = 0; r < 8; ++r)
        proj[(size_t)(rbase + r) * C4 + col] = acc[r] + bias;
}

// ---------------------------------------------------------------------------
// Kernel 6: deterministic reduction of 4096 partials -> nce scalar at d_out[0]
// ---------------------------------------------------------------------------
__global__ void k6_finalize_nce(const float* __restrict__ partials,
                                float* __restrict__ out) {
    __shared__ float buf[256];
    const int tid = threadIdx.x;
    float s = 0.f;
    for (int i = tid; i < Tdim * Wdim * (Bdim / 16); i += 256) s += partials[i];
    buf[tid] = s;
    __syncthreads();
    for (int off = 128; off > 0; off >>= 1) {
        if (tid < off) buf[tid] += buf[tid + off];
        __syncthreads();
    }
    if (tid == 0) out[0] = -buf[0] / (float)(Bdim * Tdim * Wdim);
}

// ---------------------------------------------------------------------------
// Launcher
// ---------------------------------------------------------------------------
extern "C" void kernel_launch(void* const* d_in, const int* in_sizes, int n_in,
                              void* d_out, int out_size, void* d_ws, size_t ws_size,
                              hipStream_t stream) {
    const float* features = (const float*)d_in[0];   // [B, C, L]
    const float* c_t      = (const float*)d_in[1];   // [B, W, H]
    const float* Wk_w     = (const float*)d_in[2];   // [T, C, H]
    const float* Wk_b     = (const float*)d_in[3];   // [T, C]
    const float* pw1      = (const float*)d_in[4];   // [C/2, H]
    const float* pb1      = (const float*)d_in[5];   // [C/2]
    const float* bn_gamma = (const float*)d_in[6];   // [C/2]
    const float* bn_beta  = (const float*)d_in[7];   // [C/2]
    const float* pw2      = (const float*)d_in[8];   // [C/4, C/2]
    const float* pb2      = (const float*)d_in[9];   // [C/4]
    const int*   t_sample = (const int*)  d_in[10];  // [1]

    float* out = (float*)d_out;        // out[0] = nce, out[1..] = proj [4096x64]

    // workspace layout (floats)
    float* ws_pred     = (float*)d_ws;                        // T*BW*C = 16,777,216
    float* ws_partials = ws_pred     + (size_t)Tdim * BW * Cdim;  // 4096
    float* ws_h1       = ws_partials + 4096;                  // BW*C2 = 524,288
    float* ws_scale    = ws_h1       + (size_t)BW * C2;       // 128
    float* ws_shift    = ws_scale    + C2;                    // 128

    // 1) pred = c_t @ Wk_w[t]^T + Wk_b[t]
    k1_pred_gemm<<<dim3(BW / 64, Cdim / 32, Tdim), 256, 0, stream>>>(
        c_t, Wk_w, Wk_b, ws_pred);

    // 2) fused scores + log-softmax diagonal partials
    k2_scores_nce<<<dim3(Bdim / 16, Wdim, Tdim), 256, 0, stream>>>(
        features, ws_pred, t_sample, ws_partials);

    // 3) h1 = c_t2d @ pw1^T + pb1
    k3_h1_gemm<<<dim3(BW / 64, C2 / 32, 1), 256, 0, stream>>>(
        c_t, pw1, pb1, ws_h1);

    // 4) BN train-mode stats -> fused scale/shift
    k4_bn_stats<<<dim3(C2, 1, 1), 256, 0, stream>>>(
        ws_h1, bn_gamma, bn_beta, ws_scale, ws_shift);

    // 5) proj = relu(bn(h1)) @ pw2^T + pb2  -> out+1
    k5_proj_gemm<<<dim3(BW / 64, C4 / 32, 1), 256, 0, stream>>>(
        ws_h1, ws_scale, ws_shift, pw2, pb2, out + 1);

    // 6) nce scalar -> out[0]
    k6_finalize_nce<<<dim3(1, 1, 1), 256, 0, stream>>>(ws_partials, out);
}